// RankingHead_76922864272120
// MI455X (gfx1250) — compile-verified
//
#include <hip/hip_runtime.h>

typedef __attribute__((ext_vector_type(2))) float v2f;
typedef __attribute__((ext_vector_type(8))) float v8f;

#define RK_N 4096
#define RK_D 256
#define RK_MARGIN 0.1f

// ---------------------------------------------------------------------------
// Kernel A: diag[i] = <v_i, t_i>, zero rank[] and the double accumulators.
// One wave32 per row; lane covers 8 consecutive K elements (2x float4 loads).
// ---------------------------------------------------------------------------
__global__ void rk_diag_init(const float* __restrict__ v,
                             const float* __restrict__ t,
                             float* __restrict__ diag,
                             int* __restrict__ rank,
                             double* __restrict__ acc) {
    const int lane = threadIdx.x & 31;
    const int wave = threadIdx.x >> 5;
    const int row  = blockIdx.x * 8 + wave;     // 512 blocks * 8 waves = 4096 rows

    const float4* vp = (const float4*)(v + (size_t)row * RK_D) + lane * 2;
    const float4* tp = (const float4*)(t + (size_t)row * RK_D) + lane * 2;
    float4 a0 = vp[0], a1 = vp[1];
    float4 b0 = tp[0], b1 = tp[1];
    float sum = a0.x * b0.x + a0.y * b0.y + a0.z * b0.z + a0.w * b0.w
              + a1.x * b1.x + a1.y * b1.y + a1.z * b1.z + a1.w * b1.w;

    #pragma unroll
    for (int off = 16; off > 0; off >>= 1)
        sum += __shfl_xor(sum, off);

    if (lane == 0) {
        diag[row] = sum;
        rank[row] = 0;
    }
    if (blockIdx.x == 0 && threadIdx.x < 2)
        acc[threadIdx.x] = 0.0;
}

// ---------------------------------------------------------------------------
// Kernel B: fused WMMA GEMM (s = V * T^T, fp32) + all reductions.
// Block = 256 threads = 8 waves; each wave computes one 16x16 tile of s.
// Block covers a 64-row x 32-col region: wave (wy = id&3, wx = id>>2).
// K = 256 as 64 chained V_WMMA_F32_16X16X4_F32 steps.
// ---------------------------------------------------------------------------
__global__ void rk_gemm_reduce(const float* __restrict__ v,
                               const float* __restrict__ t,
                               const float* __restrict__ diag,
                               int* __restrict__ rank,
                               double* __restrict__ acc) {
    const int lane  = threadIdx.x & 31;
    const int wave  = threadIdx.x >> 5;
    const int m     = lane & 15;     // M (for A) / N (for B) index within tile
    const int khalf = lane >> 4;     // 0: K = {0,1}(+k0); 1: K = {2,3}(+k0)

    const int rowBase = blockIdx.x * 64 + (wave & 3) * 16;
    const int colBase = blockIdx.y * 32 + (wave >> 2) * 16;

    const float* aPtr = v + (size_t)(rowBase + m) * RK_D + khalf * 2;
    const float* bPtr = t + (size_t)(colBase + m) * RK_D + khalf * 2;

    v8f cacc = {};
    for (int k0 = 0; k0 < RK_D; k0 += 32) {
        #pragma unroll
        for (int kk = 0; kk < 32; kk += 4) {
            v2f a = *(const v2f*)(aPtr + k0 + kk);
            v2f b = *(const v2f*)(bPtr + k0 + kk);
            // D = A(16x4,f32) * B(4x16,f32) + C(16x16,f32)
            cacc = __builtin_amdgcn_wmma_f32_16x16x4_f32(
                false, a, false, b, (short)0, cacc, false, false);
        }
    }

    // ---- fused epilogue -------------------------------------------------
    // C layout: VGPR vg -> global row r = rowBase + vg + 8*khalf,
    //           column c = colBase + (lane & 15).
    const int   c  = colBase + m;
    const float dC = diag[c];

    float vtSum = 0.0f, tvSum = 0.0f;
    #pragma unroll
    for (int vg = 0; vg < 8; ++vg) {
        const int   r   = rowBase + vg + 8 * khalf;
        const float s   = cacc[vg];
        const float dR  = diag[r];
        const bool  od  = (r != c);
        if (od) {
            vtSum += fmaxf(0.0f, RK_MARGIN - dR + s);
            tvSum += fmaxf(0.0f, RK_MARGIN - dC + s);
        }
        // rank increments: strictly-greater count per row.
        // ballot low 16 bits  -> lanes 0..15  -> row rowBase+vg
        // ballot high 16 bits -> lanes 16..31 -> row rowBase+8+vg
        unsigned long long mask = __ballot(od && (s > dR));
        if (lane == 0) {
            int lo = (int)__popcll(mask & 0xFFFFull);
            int hi = (int)__popcll((mask >> 16) & 0xFFFFull);
            if (lo) atomicAdd(&rank[rowBase + vg], lo);
            if (hi) atomicAdd(&rank[rowBase + 8 + vg], hi);
        }
    }

    // wave32 reduce, then LDS reduce across the 8 waves, one f64 atomic/block
    #pragma unroll
    for (int off = 16; off > 0; off >>= 1) {
        vtSum += __shfl_xor(vtSum, off);
        tvSum += __shfl_xor(tvSum, off);
    }
    __shared__ float sVt[8], sTv[8];
    if (lane == 0) { sVt[wave] = vtSum; sTv[wave] = tvSum; }
    __syncthreads();
    if (threadIdx.x == 0) {
        float va = 0.0f, ta = 0.0f;
        #pragma unroll
        for (int i = 0; i < 8; ++i) { va += sVt[i]; ta += sTv[i]; }
        atomicAdd(&acc[0], (double)va);
        atomicAdd(&acc[1], (double)ta);
    }
}

// ---------------------------------------------------------------------------
// Kernel C: finalize -> 6 fp32 outputs
// ---------------------------------------------------------------------------
__global__ void rk_finalize(const int* __restrict__ rank,
                            const double* __restrict__ acc,
                            float* __restrict__ out) {
    const int tid = threadIdx.x;
    int c1 = 0, c5 = 0, c10 = 0, rs = 0;
    for (int r = tid; r < RK_N; r += 256) {
        const int rk = rank[r];
        c1  += (rk < 1);
        c5  += (rk < 5);
        c10 += (rk < 10);
        rs  += rk;
    }
    __shared__ int s1[256], s5[256], s10[256], ss[256];
    s1[tid] = c1; s5[tid] = c5; s10[tid] = c10; ss[tid] = rs;
    __syncthreads();
    for (int s = 128; s > 0; s >>= 1) {
        if (tid < s) {
            s1[tid]  += s1[tid + s];
            s5[tid]  += s5[tid + s];
            s10[tid] += s10[tid + s];
            ss[tid]  += ss[tid + s];
        }
        __syncthreads();
    }
    if (tid == 0) {
        const double denom = (double)RK_N * (double)(RK_N - 1);
        out[0] = (float)(acc[0] / denom);            // vt_loss
        out[1] = (float)(acc[1] / denom);            // tv_loss
        out[2] = (float)s1[0]  / (float)RK_N;        // recall@1
        out[3] = (float)s5[0]  / (float)RK_N;        // recall@5
        out[4] = (float)s10[0] / (float)RK_N;        // recall@10
        out[5] = (float)ss[0]  / (float)RK_N;        // mean rank
    }
}

// ---------------------------------------------------------------------------
extern "C" void kernel_launch(void* const* d_in, const int* in_sizes, int n_in,
                              void* d_out, int out_size, void* d_ws, size_t ws_size,
                              hipStream_t stream) {
    const float* v = (const float*)d_in[0];
    const float* t = (const float*)d_in[1];
    float* out = (float*)d_out;

    char* ws      = (char*)d_ws;
    double* acc   = (double*)ws;                       // 2 doubles
    float*  diag  = (float*)(ws + 16);                 // 4096 floats
    int*    rank  = (int*)(ws + 16 + RK_N * 4);        // 4096 ints

    rk_diag_init<<<RK_N / 8, 256, 0, stream>>>(v, t, diag, rank, acc);
    rk_gemm_reduce<<<dim3(RK_N / 64, RK_N / 32), 256, 0, stream>>>(v, t, diag, rank, acc);
    rk_finalize<<<1, 256, 0, stream>>>(rank, acc, out);
}